// rnnCell_dualNet__489626271850
// MI455X (gfx1250) — compile-verified
//
#include <hip/hip_runtime.h>
#include <math.h>

#define N_NEURON 1024
#define N_PEPTIDE 16
#define N_GENES 1
#define BATCH 2048
#define SQ 32
#define STATE_SIZE (N_NEURON + N_GENES + N_NEURON + N_NEURON * N_PEPTIDE) // 18433
#define DT (1.0f / 120.0f)

typedef float v2f __attribute__((ext_vector_type(2)));
typedef float v8f __attribute__((ext_vector_type(8)));
typedef int   i4  __attribute__((vector_size(16)));   // b128 payload type

// -------------------------- async-to-LDS (CDNA5) ---------------------------
#define HAS_ASYNC_B128 0
#if defined(__has_builtin)
#  if __has_builtin(__builtin_amdgcn_global_load_async_to_lds_b128)
#    undef HAS_ASYNC_B128
#    define HAS_ASYNC_B128 1
#  endif
#endif

__device__ __forceinline__ void copy16_to_lds(const float* g, float* l) {
#if HAS_ASYNC_B128
  __builtin_amdgcn_global_load_async_to_lds_b128(
      (__attribute__((address_space(1))) i4*)(void*)g,
      (__attribute__((address_space(3))) i4*)(void*)l, 0, 0);
#else
  *reinterpret_cast<float4*>(l) = *reinterpret_cast<const float4*>(g);
#endif
}

__device__ __forceinline__ void wait_async_lds() {
#if HAS_ASYNC_B128
  asm volatile("s_wait_asynccnt 0" ::: "memory");
#endif
}

// ---------------------------------------------------------------------------
// Kernel 1: all elementwise work. One thread per (batch, neuron).
// ---------------------------------------------------------------------------
__global__ __launch_bounds__(256) void ew_kernel(
    const float* __restrict__ u,         // (B,1)
    const float* __restrict__ state,     // (B,S)
    const float* __restrict__ noise_u,   // (B,N)
    const float* __restrict__ D,         // (16,)
    const float* __restrict__ prodr,     // (16,)
    const float* __restrict__ decayr,    // (16,)
    const float* __restrict__ action,    // (16,)
    const float* __restrict__ ndec,      // (1,)
    const float* __restrict__ inlayer,   // (N,)
    float* __restrict__ firing_out,      // (B,N)
    float* __restrict__ snew)            // (B,S)
{
  int idx = blockIdx.x * blockDim.x + threadIdx.x;
  int b = idx >> 10;
  int n = idx & (N_NEURON - 1);
  if (b >= BATCH) return;

  const float* st = state + (size_t)b * STATE_SIZE;
  float*       sn = snew  + (size_t)b * STATE_SIZE;

  float abl = st[n];
  float pc2 = st[N_NEURON];
  float nrn = st[N_NEURON + N_GENES + n];
  float noi = noise_u[b * N_NEURON + n];

  float nm = nrn * abl;                       // masked neuron
  float f  = fmaxf(nrn, 0.0f) * abl;          // relu * ablate
  f = -(f + 0.01f) * log1pf(-noi);
  f = fminf(fmaxf(f, 0.0f), 10.0f);

  firing_out[b * N_NEURON + n] = f;
  sn[n] = abl;
  if (n == 0) sn[N_NEURON] = pc2;

  // periodic 32x32 Laplacian neighbor indices
  int r = n >> 5, c = n & (SQ - 1);
  int n_up = (((r + SQ - 1) & (SQ - 1)) << 5) | c;
  int n_dn = (((r + 1) & (SQ - 1)) << 5) | c;
  int n_lf = (r << 5) | ((c + SQ - 1) & (SQ - 1));
  int n_rt = (r << 5) | ((c + 1) & (SQ - 1));

  const int pbase = 2 * N_NEURON + N_GENES;
  const float* pep = st + pbase + n    * N_PEPTIDE;
  const float* pu  = st + pbase + n_up * N_PEPTIDE;
  const float* pd  = st + pbase + n_dn * N_PEPTIDE;
  const float* pl  = st + pbase + n_lf * N_PEPTIDE;
  const float* pr  = st + pbase + n_rt * N_PEPTIDE;
  float*       pn  = sn + pbase + n    * N_PEPTIDE;

  float actsum = 0.0f;
#pragma unroll
  for (int p = 0; p < N_PEPTIDE; ++p) {
    float pv  = pep[p];
    float lap = pu[p] + pd[p] + pl[p] + pr[p] - 4.0f * pv;
    float dp  = fabsf(prodr[p]) * f - fabsf(decayr[p]) * pv + fabsf(D[p]) * lap;
    pn[p]     = pv + dp * DT;
    actsum   += pv * action[p];
  }

  float partial = actsum * pc2 - nm * fabsf(ndec[0]) + inlayer[n] * u[b];
  sn[N_NEURON + N_GENES + n] = nm + partial * DT; // "pre", finished by GEMM epilogue
}

// ---------------------------------------------------------------------------
// Kernel 2: firing @ synapse via V_WMMA_F32_16X16X4_F32, fused epilogue:
//   neuron_new = (pre + DT * syn) * ablate
//
// 256 threads = 8 waves; tile 128(M) x 64(N); each wave owns 32x32
// (4 f32 accumulators). BK = 32, DOUBLE-BUFFERED:
//   iter t: commit B(t) regs -> LDS, s_wait_asynccnt 0, barrier,
//           issue async A(t+1) + load B(t+1) regs, then 32 x v_wmma on buf t.
// A staged row-major (stride 36, conflict-free, 16B rows) via
// GLOBAL_LOAD_ASYNC_TO_LDS_B128. B staged K-pair-packed:
//   Bp[kp][n] = { B[2kp][n], B[2kp+1][n] }  (pair-row stride 160 words)
// so every fragment operand is one aligned ds_load_b64.
// ---------------------------------------------------------------------------
#define BM 128
#define BN 64
#define BK 32
#define ASS 36    // A LDS stride (words)
#define BPS 160   // B pair-row stride (words); 160 mod 64 = 32 -> halves disjoint
#define T_TILES (N_NEURON / BK)

__global__ __launch_bounds__(256) void gemm_kernel(
    const float* __restrict__ firing,   // (B, N)
    const float* __restrict__ synm,     // (N, N)
    float* __restrict__ snew)           // (B, S)
{
  __shared__ float As[2][BM * ASS];           // 2 x 18432 B
  __shared__ float Bp[2][(BK / 2) * BPS];     // 2 x 10240 B

  const int tid  = threadIdx.x;
  const int lane = tid & 31;
  const int wave = tid >> 5;
  const int wm   = wave >> 1;     // 0..3 : 32-row M subtile
  const int wn   = wave & 1;      // 0..1 : 32-col N subtile
  const int m0   = blockIdx.y * BM;
  const int n0   = blockIdx.x * BN;
  const int half = lane >> 4;     // K sub-pair / C row-half selector
  const int lm   = lane & 15;

  // B staging thread map: one (k-pair, col4) slot per thread
  const int skp   = tid >> 4;     // 0..15
  const int scol4 = tid & 15;     // 0..15

  v8f c00 = {}, c01 = {}, c10 = {}, c11 = {};

  // ---- prologue: stage tile 0 ----
#pragma unroll
  for (int i = 0; i < 4; ++i) {
    int slot = i * 256 + tid;
    int row  = slot >> 3;
    int col4 = slot & 7;
    copy16_to_lds(firing + (size_t)(m0 + row) * N_NEURON + col4 * 4,
                  &As[0][row * ASS + col4 * 4]);
  }
  const float* gB0 = synm + (size_t)(2 * skp) * N_NEURON + n0 + scol4 * 4;
  float4 r0 = *reinterpret_cast<const float4*>(gB0);
  float4 r1 = *reinterpret_cast<const float4*>(gB0 + N_NEURON);

  for (int t = 0; t < T_TILES; ++t) {
    const int cur = t & 1;

    // commit B(t) registers into LDS (K-pair-packed)
    {
      float a0v[4] = {r0.x, r0.y, r0.z, r0.w};
      float a1v[4] = {r1.x, r1.y, r1.z, r1.w};
#pragma unroll
      for (int j = 0; j < 4; ++j) {
        *reinterpret_cast<float2*>(&Bp[cur][skp * BPS + (scol4 * 4 + j) * 2]) =
            make_float2(a0v[j], a1v[j]);
      }
    }
    wait_async_lds();       // only tile-t asyncs are outstanding here
    __syncthreads();        // staging of t visible; compute of t-1 finished

    // prefetch tile t+1 into the other buffer while computing tile t
    if (t + 1 < T_TILES) {
      const int kt = (t + 1) * BK;
#pragma unroll
      for (int i = 0; i < 4; ++i) {
        int slot = i * 256 + tid;
        int row  = slot >> 3;
        int col4 = slot & 7;
        copy16_to_lds(firing + (size_t)(m0 + row) * N_NEURON + kt + col4 * 4,
                      &As[cur ^ 1][row * ASS + col4 * 4]);
      }
      const float* gB = synm + (size_t)(kt + 2 * skp) * N_NEURON + n0 + scol4 * 4;
      r0 = *reinterpret_cast<const float4*>(gB);
      r1 = *reinterpret_cast<const float4*>(gB + N_NEURON);
    }

#pragma unroll
    for (int k = 0; k < BK; k += 4) {
      const int kp = (k >> 1) + half;   // half=0 -> {k,k+1}; half=1 -> {k+2,k+3}
      v2f a0 = *reinterpret_cast<const v2f*>(&As[cur][(wm * 32 + lm)      * ASS + k + half * 2]);
      v2f a1 = *reinterpret_cast<const v2f*>(&As[cur][(wm * 32 + 16 + lm) * ASS + k + half * 2]);
      v2f b0 = *reinterpret_cast<const v2f*>(&Bp[cur][kp * BPS + (wn * 32 + lm) * 2]);
      v2f b1 = *reinterpret_cast<const v2f*>(&Bp[cur][kp * BPS + (wn * 32 + 16 + lm) * 2]);

      c00 = __builtin_amdgcn_wmma_f32_16x16x4_f32(false, a0, false, b0, (short)0, c00, false, false);
      c01 = __builtin_amdgcn_wmma_f32_16x16x4_f32(false, a0, false, b1, (short)0, c01, false, false);
      c10 = __builtin_amdgcn_wmma_f32_16x16x4_f32(false, a1, false, b0, (short)0, c10, false, false);
      c11 = __builtin_amdgcn_wmma_f32_16x16x4_f32(false, a1, false, b1, (short)0, c11, false, false);
    }
    // no trailing barrier: next iteration's barrier fences buffer reuse
  }

  // Epilogue: C layout — VGPR i: lanes 0-15 = row i, lanes 16-31 = row i+8
  const int nA = n0 + wn * 32 + lm;
  const int nB = nA + 16;
#pragma unroll
  for (int i = 0; i < 8; ++i) {
    int mA = m0 + wm * 32 + half * 8 + i;
    int mB = mA + 16;
    float* snA = snew + (size_t)mA * STATE_SIZE;
    float* snB = snew + (size_t)mB * STATE_SIZE;
    snA[N_NEURON + N_GENES + nA] =
        (snA[N_NEURON + N_GENES + nA] + DT * c00[i]) * snA[nA];
    snA[N_NEURON + N_GENES + nB] =
        (snA[N_NEURON + N_GENES + nB] + DT * c01[i]) * snA[nB];
    snB[N_NEURON + N_GENES + nA] =
        (snB[N_NEURON + N_GENES + nA] + DT * c10[i]) * snB[nA];
    snB[N_NEURON + N_GENES + nB] =
        (snB[N_NEURON + N_GENES + nB] + DT * c11[i]) * snB[nB];
  }
}

// ---------------------------------------------------------------------------
extern "C" void kernel_launch(void* const* d_in, const int* in_sizes, int n_in,
                              void* d_out, int out_size, void* d_ws, size_t ws_size,
                              hipStream_t stream) {
  const float* u       = (const float*)d_in[0];
  const float* state   = (const float*)d_in[1];
  const float* noise   = (const float*)d_in[2];
  const float* D       = (const float*)d_in[3];
  const float* prodr   = (const float*)d_in[4];
  const float* decayr  = (const float*)d_in[5];
  const float* action  = (const float*)d_in[6];
  const float* synm    = (const float*)d_in[7];
  const float* ndec    = (const float*)d_in[8];
  const float* inlayer = (const float*)d_in[9];

  float* firing_out = (float*)d_out;                              // (B, N)
  float* snew       = firing_out + (size_t)BATCH * N_NEURON;      // (B, S)

  ew_kernel<<<(BATCH * N_NEURON) / 256, 256, 0, stream>>>(
      u, state, noise, D, prodr, decayr, action, ndec, inlayer,
      firing_out, snew);

  dim3 grid(N_NEURON / BN, BATCH / BM);
  gemm_kernel<<<grid, 256, 0, stream>>>(firing_out, synm, snew);
}